// QANet_15101105013510
// MI455X (gfx1250) — compile-verified
//
// QANet forward for MI455X (gfx1250, wave32). All matmuls (pointwise convs,
// highway, attention projections, attention scores/head, CQ attention, FF)
// go through one wave-level WMMA GEMM using v_wmma_f32_16x16x4_f32 (f32
// matches the reference precision; the net is latency/L2-bound — ~34 GFLOP
// total, activations ~MBs vs 192MB L2 — so fp32 matrix ops are the right
// ceiling, not fp8). A wave computes an MBx16-row x 16-col macro-tile
// (MB templated: 4 for large-M GEMMs, 1 for the 16-row projections); the
// B fragment is loaded once per K-step and reused by MB unconditional WMMAs
// (accumulators never conditionally assigned -> no phi/copy storms). Tail
// macro-tiles clamp the A row index so loads stay in-bounds; only the
// epilogue is guarded. Edge shapes (L=50/K=50) use clamped loads + value
// selects instead of divergent control flow.
#include <hip/hip_runtime.h>
#include <hip/hip_bf16.h>

typedef __attribute__((ext_vector_type(2))) float v2f;
typedef __attribute__((ext_vector_type(8))) float v8f;

#define BIGMOD (1 << 30)

// ---------------------------------------------------------------------------
// Generic batched WMMA GEMM:  O[z][m][n] = act(alpha * sum_k A[m,k]*X[k,n]
//                                             + bias[m] + rowBias[z,m]) + resid
//   A offset   = z * aStride            (TRA: A[k*lda+m])
//   X offset   = (z % xMod) * xStride   (TRX: X[n*ldx+k])
//   O offset   = (z % oMod)*oStr1 + (z / oMod)*oStr2 ; row-major (m,n), ld=N
// ---------------------------------------------------------------------------
struct GemmArgs {
  const float *A, *X;
  float *O;
  const float *bias, *rowBias, *resid;
  int M, N, K, lda, ldx, transA, transX;
  long aStride;
  int xMod; long xStride;
  int oMod; long oStr1, oStr2;
  float alpha;
  int act; // 0 none, 1 relu, 2 sigmoid
};

template <int MB, int TRA, int TRX, bool CHECK>
__global__ __launch_bounds__(32) void wmma_gemm_t(GemmArgs g) {
  const int z    = blockIdx.z;
  const int lane = threadIdx.x;
  const int half = lane >> 4;     // 0: lanes 0-15, 1: lanes 16-31
  const int l16  = lane & 15;
  const int m0   = blockIdx.y * (16 * MB);
  const int n0   = blockIdx.x << 4;

  const float* A = g.A + (long)z * g.aStride;
  const float* X = g.X + (long)(z % g.xMod) * g.xStride;
  const long ooff = (long)(z % g.oMod) * g.oStr1 + (long)(z / g.oMod) * g.oStr2;

  const int bn = n0 + l16;    // B col this lane supplies
  const int kh = half * 2;    // lanes 16-31 hold K = k0+2, k0+3

  v8f acc[MB] = {};

  if (!CHECK) {
    // Fast path: M%16==0, N%16==0, K%4==0. Branch-free, pointer-bumped.
    // Tail macro-tile: clamp row index so loads stay in-bounds; the garbage
    // subtiles are simply not stored in the epilogue.
    long ai[MB];
#pragma unroll
    for (int j = 0; j < MB; ++j) {
      int am = m0 + 16 * j + l16;
      if (am > g.M - 1) am = g.M - 1;
      ai[j] = TRA ? (long)kh * g.lda + am : (long)am * g.lda + kh;
    }
    long xi = TRX ? (long)bn * g.ldx + kh : (long)kh * g.ldx + bn;
    const long aStep = TRA ? 4L * g.lda : 4L;
    const long xStep = TRX ? 4L : 4L * g.ldx;
    const int iters = g.K >> 2;
#pragma unroll 2
    for (int t = 0; t < iters; ++t) {
      v2f bv;
      if (TRX) {
        const float2 tmp = *(const float2*)(X + xi);   // contiguous, 8B aligned
        bv.x = tmp.x; bv.y = tmp.y;
      } else {
        bv.x = X[xi]; bv.y = X[xi + g.ldx];
      }
#pragma unroll
      for (int j = 0; j < MB; ++j) {
        v2f av;
        if (TRA) {
          av.x = A[ai[j]]; av.y = A[ai[j] + g.lda];
        } else {
          const float2 t2 = *(const float2*)(A + ai[j]);  // 8B aligned
          av.x = t2.x; av.y = t2.y;
        }
        acc[j] = __builtin_amdgcn_wmma_f32_16x16x4_f32(false, av, false, bv, (short)0,
                                                       acc[j], false, false);
      }
#pragma unroll
      for (int j = 0; j < MB; ++j) ai[j] += aStep;
      xi += xStep;
    }
  } else {
    // Edge path (L=50 / K=50 shapes): clamped loads + value selects, no
    // divergence, WMMA unconditional.
    const int bok = bn < g.N;
    const int bnc = bok ? bn : g.N - 1;
    for (int k0 = 0; k0 < g.K; k0 += 4) {
      const int ka = k0 + kh;
      const int k0ok = ka < g.K;
      const int k1ok = ka + 1 < g.K;
      const int ka0 = k0ok ? ka : g.K - 1;
      const int ka1 = k1ok ? ka + 1 : g.K - 1;
      float b0 = TRX ? X[(long)bnc * g.ldx + ka0] : X[(long)ka0 * g.ldx + bnc];
      float b1 = TRX ? X[(long)bnc * g.ldx + ka1] : X[(long)ka1 * g.ldx + bnc];
      b0 = (bok && k0ok) ? b0 : 0.f;
      b1 = (bok && k1ok) ? b1 : 0.f;
      v2f bv = {b0, b1};
#pragma unroll
      for (int j = 0; j < MB; ++j) {
        const int am  = m0 + j * 16 + l16;
        const int aok = am < g.M;
        const int amc = aok ? am : g.M - 1;
        float a0 = TRA ? A[(long)ka0 * g.lda + amc] : A[(long)amc * g.lda + ka0];
        float a1 = TRA ? A[(long)ka1 * g.lda + amc] : A[(long)amc * g.lda + ka1];
        a0 = (aok && k0ok) ? a0 : 0.f;
        a1 = (aok && k1ok) ? a1 : 0.f;
        v2f av = {a0, a1};
        acc[j] = __builtin_amdgcn_wmma_f32_16x16x4_f32(false, av, false, bv, (short)0,
                                                       acc[j], false, false);
      }
    }
  }

#pragma unroll
  for (int j = 0; j < MB; ++j) {
    if (MB == 1 || m0 + j * 16 < g.M) {                 // uniform subtile gate
#pragma unroll
      for (int r = 0; r < 8; ++r) {
        const int mm = m0 + j * 16 + half * 8 + r;      // C/D: VGPR r -> row r / 8+r
        const int nn = n0 + l16;
        if (!CHECK || (mm < g.M && nn < g.N)) {
          float v = acc[j][r] * g.alpha;
          if (g.bias)    v += g.bias[mm];
          if (g.rowBias) v += g.rowBias[(long)z * g.M + mm];
          if (g.act == 1)      v = v > 0.f ? v : 0.f;
          else if (g.act == 2) v = 1.f / (1.f + expf(-v));
          const long oi = ooff + (long)mm * g.N + nn;
          if (g.resid) v += g.resid[oi];
          g.O[oi] = v;
        }
      }
    }
  }
}

// ---------------------------------------------------------------------------
// Elementwise / reduction kernels
// ---------------------------------------------------------------------------
__global__ __launch_bounds__(256) void pe_k(float* pe, int L, int n) {
  int i = blockIdx.x * 256 + threadIdx.x;
  if (i >= n) return;
  int d = i / L, pos = i % L;
  float freq, ph;
  if ((d & 1) == 0) { freq = powf(10000.f, -(float)d / 128.f);       ph = 0.f; }
  else              { freq = -powf(10000.f, -(float)(d - 1) / 128.f); ph = 1.57079632679f; }
  pe[i] = sinf((float)pos * freq + ph);
}

__global__ __launch_bounds__(256) void add_pe_k(const float* x, const float* pe, float* o,
                                                int DL, int n) {
  int i = blockIdx.x * 256 + threadIdx.x;
  if (i >= n) return;
  o[i] = x[i] + pe[i % DL];
}

__global__ __launch_bounds__(256) void copy_k(const float* x, float* o, int n) {
  int i = blockIdx.x * 256 + threadIdx.x;
  if (i >= n) return;
  o[i] = x[i];
}

// layernorm over the full (D,L) slab of each batch sample, affine w,b of (D,L)
__global__ __launch_bounds__(256) void layernorm_k(const float* x, const float* w,
                                                   const float* b, float* o, int DL) {
  const int bi = blockIdx.x;
  const float* xb = x + (long)bi * DL;
  float s = 0.f, s2 = 0.f;
  for (int i = threadIdx.x; i < DL; i += 256) { float v = xb[i]; s += v; s2 += v * v; }
  __shared__ float sh0[256], sh1[256];
  sh0[threadIdx.x] = s; sh1[threadIdx.x] = s2; __syncthreads();
  for (int st = 128; st > 0; st >>= 1) {
    if (threadIdx.x < st) { sh0[threadIdx.x] += sh0[threadIdx.x + st];
                            sh1[threadIdx.x] += sh1[threadIdx.x + st]; }
    __syncthreads();
  }
  const float mean = sh0[0] / (float)DL;
  const float var  = sh1[0] / (float)DL - mean * mean;
  const float rs   = rsqrtf(var + 1e-5f);
  for (int i = threadIdx.x; i < DL; i += 256)
    o[(long)bi * DL + i] = (xb[i] - mean) * rs * w[i] + b[i];
}

// softmax over rows (dim R) for each column of a (R x C) matrix, batch = gridDim.y
__global__ __launch_bounds__(256) void colsoftmax_k(float* S, int R, int C) {
  const long base = (long)blockIdx.y * R * C;
  const int col = blockIdx.x, tid = threadIdx.x;
  __shared__ float sh[256];
  float m = -3.4e38f;
  for (int r = tid; r < R; r += 256) m = fmaxf(m, S[base + (long)r * C + col]);
  sh[tid] = m; __syncthreads();
  for (int s = 128; s > 0; s >>= 1) { if (tid < s) sh[tid] = fmaxf(sh[tid], sh[tid + s]); __syncthreads(); }
  m = sh[0]; __syncthreads();
  float sum = 0.f;
  for (int r = tid; r < R; r += 256) sum += expf(S[base + (long)r * C + col] - m);
  sh[tid] = sum; __syncthreads();
  for (int s = 128; s > 0; s >>= 1) { if (tid < s) sh[tid] += sh[tid + s]; __syncthreads(); }
  const float inv = 1.f / sh[0];
  for (int r = tid; r < R; r += 256) {
    long idx = base + (long)r * C + col;
    S[idx] = expf(S[idx] - m) * inv;
  }
}

// softmax over the columns of row blockIdx.x (batch = gridDim.y)
__global__ __launch_bounds__(256) void rowsoftmax_k(float* S, int R, int C) {
  const long base = ((long)blockIdx.y * R + blockIdx.x) * C;
  const int tid = threadIdx.x;
  __shared__ float sh[256];
  float m = -3.4e38f;
  for (int c = tid; c < C; c += 256) m = fmaxf(m, S[base + c]);
  sh[tid] = m; __syncthreads();
  for (int s = 128; s > 0; s >>= 1) { if (tid < s) sh[tid] = fmaxf(sh[tid], sh[tid + s]); __syncthreads(); }
  m = sh[0]; __syncthreads();
  float sum = 0.f;
  for (int c = tid; c < C; c += 256) sum += expf(S[base + c] - m);
  sh[tid] = sum; __syncthreads();
  for (int s = 128; s > 0; s >>= 1) { if (tid < s) sh[tid] += sh[tid + s]; __syncthreads(); }
  const float inv = 1.f / sh[0];
  for (int c = tid; c < C; c += 256) S[base + c] = expf(S[base + c] - m) * inv;
}

// depthwise 1D conv, weights w[c*k+j], zero pad k/2, optional per-channel bias
__global__ __launch_bounds__(256) void dwconv1d_k(const float* x, const float* w,
                                                  const float* bias, float* o,
                                                  int C, int L, int k, int n) {
  int i = blockIdx.x * 256 + threadIdx.x;
  if (i >= n) return;
  const int l = i % L;
  const int c = (i / L) % C;
  const long rowbase = (long)i - l;
  float acc = bias ? bias[c] : 0.f;
  const int h = k >> 1;
  for (int j = 0; j < k; ++j) {
    int ll = l + j - h;
    if (ll < 0 || ll >= L) continue;
    acc += x[rowbase + ll] * w[c * k + j];
  }
  o[i] = acc;
}

// depthwise 5x5 conv on (B,64,L,16), pad 2, per-channel bias
__global__ __launch_bounds__(256) void dwconv2d_k(const float* x, const float* w,
                                                  const float* bias, float* o,
                                                  int L, int n) {
  int i = blockIdx.x * 256 + threadIdx.x;
  if (i >= n) return;
  const int ww = i & 15;
  const int h  = (i >> 4) % L;
  const int c  = (i / (16 * L)) % 64;
  const long pbase = (long)i - ((long)h << 4) - ww;
  float acc = bias[c];
  const float* wk = w + c * 25;
  for (int dy = 0; dy < 5; ++dy) {
    int hy = h + dy - 2;
    if (hy < 0 || hy >= L) continue;
    for (int dx = 0; dx < 5; ++dx) {
      int wx = ww + dx - 2;
      if (wx < 0 || wx >= 16) continue;
      acc += x[pbase + (long)hy * 16 + wx] * wk[dy * 5 + dx];
    }
  }
  o[i] = acc;
}

// cc_t[b,ch,c,w] = char_emb[Ccid[b,c,w], ch]  (n = B*64*L*16)
__global__ __launch_bounds__(256) void gather_char_k(const int* cid, const float* emb,
                                                     float* out, int L, int n) {
  int i = blockIdx.x * 256 + threadIdx.x;
  if (i >= n) return;
  const int w  = i & 15;
  const int c  = (i >> 4) % L;
  const int ch = (i / (16 * L)) % 64;
  const int b  = i / (64 * 16 * L);
  const int id = cid[((long)b * L + c) * 16 + w];
  out[i] = emb[(long)id * 64 + ch];
}

// embin rows 128..427 <- word_emb[Cwid]   (n = B*300*L)
__global__ __launch_bounds__(256) void gather_word_k(const int* wid, const float* emb,
                                                     float* embin, int L, int n) {
  int i = blockIdx.x * 256 + threadIdx.x;
  if (i >= n) return;
  const int l  = i % L;
  const int ch = (i / L) % 300;
  const int b  = i / (300 * L);
  const int id = wid[(long)b * L + l];
  embin[((long)b * 428 + 128 + ch) * L + l] = emb[(long)id * 300 + ch];
}

// embin rows 0..127 <- max_w relu(pconv[b,ch,c,w])   (n = B*128*L)
__global__ __launch_bounds__(256) void relumax_k(const float* p, float* embin, int L, int n) {
  int i = blockIdx.x * 256 + threadIdx.x;
  if (i >= n) return;
  const int c = i % L;
  const long bch = i / L;
  const int b = (int)(bch / 128);
  const int ch = (int)(bch % 128);
  const long base = bch * (long)L * 16 + (long)c * 16;
  float m = 0.f;                                  // relu floor
  for (int w = 0; w < 16; ++w) m = fmaxf(m, p[base + w]);
  embin[((long)b * 428 + ch) * L + c] = m;
}

__global__ __launch_bounds__(256) void highway_k(float* x, const float* g, const float* nl,
                                                 int n) {
  int i = blockIdx.x * 256 + threadIdx.x;
  if (i >= n) return;
  const float gg = g[i];
  x[i] = gg * nl[i] + (1.f - gg) * x[i];
}

// E[b,c,d] = W1 + W3*C ; W row = cqW[(b*400+c)*384 ...]
__global__ __launch_bounds__(256) void cq_E_k(const float* W, const float* Ce, float* E,
                                              int n) {
  int i = blockIdx.x * 256 + threadIdx.x;
  if (i >= n) return;
  const int d = i % 128;
  const long row = i / 128;                 // b*400 + c
  const int b = (int)(row / 400), c = (int)(row % 400);
  const float cv = Ce[((long)b * 128 + d) * 400 + c];
  const float* Wr = W + row * 384;
  E[i] = Wr[d] + Wr[256 + d] * cv;
}

// rowterm[b*400+c] = sum_d W2[b,c,d]*C[b,c,d]
__global__ __launch_bounds__(128) void cq_rowterm_k(const float* W, const float* Ce,
                                                    float* rt) {
  const int b = blockIdx.x / 400, c = blockIdx.x % 400;
  const int d = threadIdx.x;
  const float* Wr = W + (long)blockIdx.x * 384;
  float v = Wr[128 + d] * Ce[((long)b * 128 + d) * 400 + c];
  __shared__ float sh[128];
  sh[d] = v; __syncthreads();
  for (int s = 64; s > 0; s >>= 1) { if (d < s) sh[d] += sh[d + s]; __syncthreads(); }
  if (d == 0) rt[blockIdx.x] = sh[0];
}

// X4 = concat([C, A, C*A, C*Bm]) in (B,512,400)
__global__ __launch_bounds__(256) void cq_concat_k(const float* Ce, const float* At,
                                                   const float* Bt, float* X4, int n) {
  int i = blockIdx.x * 256 + threadIdx.x;
  if (i >= n) return;
  const int l = i % 400;
  const int r = (i / 400) % 512;
  const int b = i / (512 * 400);
  const int rr = r & 127, sec = r >> 7;
  const long ci = ((long)b * 128 + rr) * 400 + l;
  const float cv = Ce[ci];
  float v;
  if (sec == 0)      v = cv;
  else if (sec == 1) v = At[ci];
  else if (sec == 2) v = cv * At[ci];
  else               v = cv * Bt[ci];
  X4[i] = v;
}

// Y[b,l] = sum_d w[b,d]*Ma[b,d,l] + w[b,128+d]*Mb[b,d,l]
__global__ __launch_bounds__(256) void ptr_score_k(const float* w, const float* Ma,
                                                   const float* Mb, float* out, int L,
                                                   int n) {
  int i = blockIdx.x * 256 + threadIdx.x;
  if (i >= n) return;
  const int l = i % L, b = i / L;
  const float* wb = w + b * 256;
  float s = 0.f;
  for (int d = 0; d < 128; ++d) {
    s += wb[d]       * Ma[((long)b * 128 + d) * L + l];
    s += wb[128 + d] * Mb[((long)b * 128 + d) * L + l];
  }
  out[i] = s;
}

// ---------------------------------------------------------------------------
// Host-side orchestration
// ---------------------------------------------------------------------------
static GemmArgs GA(const float* A, int lda, const float* X, int ldx, float* O,
                   int M, int N, int K) {
  GemmArgs g{};
  g.A = A; g.X = X; g.O = O; g.M = M; g.N = N; g.K = K; g.lda = lda; g.ldx = ldx;
  g.transA = 0; g.transX = 0; g.aStride = 0;
  g.xMod = BIGMOD; g.xStride = 0;
  g.oMod = BIGMOD; g.oStr1 = 0; g.oStr2 = 0;
  g.alpha = 1.f; g.bias = nullptr; g.rowBias = nullptr; g.resid = nullptr; g.act = 0;
  return g;
}

template <int TRA, int TRX>
static void gemm_launch(hipStream_t st, const GemmArgs& g, int Z, bool chk) {
  if (g.M >= 64) {
    dim3 grid((g.N + 15) / 16, (g.M + 63) / 64, Z);
    if (chk) wmma_gemm_t<4, TRA, TRX, true ><<<grid, 32, 0, st>>>(g);
    else     wmma_gemm_t<4, TRA, TRX, false><<<grid, 32, 0, st>>>(g);
  } else {
    dim3 grid((g.N + 15) / 16, (g.M + 15) / 16, Z);
    if (chk) wmma_gemm_t<1, TRA, TRX, true ><<<grid, 32, 0, st>>>(g);
    else     wmma_gemm_t<1, TRA, TRX, false><<<grid, 32, 0, st>>>(g);
  }
}

static void gemm(hipStream_t st, const GemmArgs& g, int Z) {
  const bool chk = ((g.M & 15) != 0) || ((g.N & 15) != 0) || ((g.K & 3) != 0);
  if (g.transA) { if (g.transX) gemm_launch<1,1>(st, g, Z, chk);
                  else          gemm_launch<1,0>(st, g, Z, chk); }
  else          { if (g.transX) gemm_launch<0,1>(st, g, Z, chk);
                  else          gemm_launch<0,0>(st, g, Z, chk); }
}

struct EncP {
  const float *dw, *pw, *Wq, *Wk, *Wv, *Wo, *W, *lnw, *lnb;
  int conv_num, k;
};

static void enc_block(hipStream_t st, float* dst, const float* src, int L, const EncP& p,
                      const float* pe, float* bufA, float* bufB, float* lnbuf, float* dwbuf,
                      float* qb, float* kb, float* vb, float* hb, float* S) {
  const int DL = 128 * L;
  const int n = 8 * DL;
  const int blk = (n + 255) / 256;
  add_pe_k<<<blk, 256, 0, st>>>(src, pe, bufA, DL, n);
  float* cur = bufA;
  for (int i = 0; i < p.conv_num; ++i) {
    layernorm_k<<<8, 256, 0, st>>>(cur, p.lnw, p.lnb, lnbuf, DL);
    dwconv1d_k<<<blk, 256, 0, st>>>(lnbuf, p.dw + (long)i * 128 * p.k, nullptr, dwbuf,
                                    128, L, p.k, n);
    float* nxt = (cur == bufA) ? bufB : bufA;
    GemmArgs g = GA(p.pw + (long)i * 16384, 128, dwbuf, L, nxt, 128, L, 128);
    g.xStride = DL; g.oStr1 = DL; g.resid = cur; g.act = 1;
    gemm(st, g, 8);
    cur = nxt;
  }
  // ---- self-attention (8 heads, per-batch weights) ----
  layernorm_k<<<8, 256, 0, st>>>(cur, p.lnw, p.lnb, lnbuf, DL);
  { GemmArgs g = GA(p.Wq, 128, lnbuf, L, qb, 16, L, 128);
    g.aStride = 2048; g.xMod = 8; g.xStride = DL; g.oStr1 = (long)16 * L; gemm(st, g, 64); }
  { GemmArgs g = GA(p.Wk, 128, lnbuf, L, kb, 16, L, 128);
    g.aStride = 2048; g.xMod = 8; g.xStride = DL; g.oStr1 = (long)16 * L; gemm(st, g, 64); }
  { GemmArgs g = GA(p.Wv, 128, lnbuf, L, vb, 16, L, 128);
    g.aStride = 2048; g.xMod = 8; g.xStride = DL; g.oStr1 = (long)16 * L; gemm(st, g, 64); }
  // s[l,m] = 0.25 * sum_k q[k,l] k[k,m]
  { GemmArgs g = GA(qb, L, kb, L, S, L, L, 16);
    g.transA = 1; g.aStride = (long)16 * L; g.xStride = (long)16 * L;
    g.oStr1 = (long)L * L; g.alpha = 0.25f; gemm(st, g, 64); }
  colsoftmax_k<<<dim3(L, 64), 256, 0, st>>>(S, L, L);       // softmax over l (rows)
  // head[v,m] = sum_l v[v,l] p[l,m]  -> packed into (B,128,L)
  { GemmArgs g = GA(vb, L, S, L, hb, 16, L, L);
    g.aStride = (long)16 * L; g.xStride = (long)L * L;
    g.oMod = 8; g.oStr1 = DL; g.oStr2 = (long)16 * L; gemm(st, g, 64); }
  float* nxt = (cur == bufA) ? bufB : bufA;
  { GemmArgs g = GA(p.Wo, 128, hb, L, nxt, 128, L, 128);
    g.aStride = 16384; g.xStride = DL; g.oStr1 = DL; g.resid = cur; gemm(st, g, 8); }
  cur = nxt;
  // ---- feed-forward ----
  layernorm_k<<<8, 256, 0, st>>>(cur, p.lnw, p.lnb, lnbuf, DL);
  { GemmArgs g = GA(p.W, 128, lnbuf, L, dst, 128, L, 128);
    g.aStride = 16384; g.xStride = DL; g.oStr1 = DL; g.resid = cur; g.act = 1;
    gemm(st, g, 8); }
}

extern "C" void kernel_launch(void* const* d_in, const int* in_sizes, int n_in,
                              void* d_out, int out_size, void* d_ws, size_t ws_size,
                              hipStream_t stream) {
  (void)in_sizes; (void)n_in; (void)out_size; (void)ws_size;
  hipStream_t st = stream;

  // ---- inputs (setup_inputs dict order, params flattened recursively) ----
  int ii = 0;
  const int*   Cwid = (const int*)d_in[ii++];        // (8,400)
  const int*   Ccid = (const int*)d_in[ii++];        // (8,400,16)
  const int*   Qwid = (const int*)d_in[ii++];        // (8,50)
  const int*   Qcid = (const int*)d_in[ii++];        // (8,50,16)
  const float* char_emb = (const float*)d_in[ii++];  // (1000,64)
  const float* word_emb = (const float*)d_in[ii++];  // (1000,300)
  // emb
  const float* c2_dw  = (const float*)d_in[ii++];    // (64,1,5,5)
  const float* c2_dwb = (const float*)d_in[ii++];
  const float* c2_pw  = (const float*)d_in[ii++];    // (128,64,1,1)
  const float* c2_pwb = (const float*)d_in[ii++];
  const float* c1_dw  = (const float*)d_in[ii++];    // (428,1,5)
  const float* c1_dwb = (const float*)d_in[ii++];
  const float* c1_pw  = (const float*)d_in[ii++];    // (128,428,1)
  const float* c1_pwb = (const float*)d_in[ii++];
  const float* lin_w  = (const float*)d_in[ii++];    // (2,128,128)
  const float* lin_b  = (const float*)d_in[ii++];
  const float* gate_w = (const float*)d_in[ii++];
  const float* gate_b = (const float*)d_in[ii++];
  // c_enc / q_enc / m_enc: {dw, pw, Wq, Wk, Wv, Wo, W, ln_w, ln_b}
  EncP cEnc{}, qEnc{}, mEnc{};
  EncP* encs[3] = {&cEnc, &qEnc, &mEnc};
  for (int e = 0; e < 3; ++e) {
    EncP& p = *encs[e];
    p.dw  = (const float*)d_in[ii++];
    p.pw  = (const float*)d_in[ii++];
    p.Wq  = (const float*)d_in[ii++];
    p.Wk  = (const float*)d_in[ii++];
    p.Wv  = (const float*)d_in[ii++];
    p.Wo  = (const float*)d_in[ii++];
    p.W   = (const float*)d_in[ii++];
    p.lnw = (const float*)d_in[ii++];
    p.lnb = (const float*)d_in[ii++];
  }
  cEnc.conv_num = 4; cEnc.k = 7;
  qEnc.conv_num = 4; qEnc.k = 7;
  mEnc.conv_num = 2; mEnc.k = 5;
  const float* cqW    = (const float*)d_in[ii++];    // (3200,1,384)
  const float* rs_dw  = (const float*)d_in[ii++];    // (512,1,5)
  const float* rs_pw  = (const float*)d_in[ii++];    // (128,512,1)
  const float* ptr_W1 = (const float*)d_in[ii++];    // (8,1,256)
  const float* ptr_W2 = (const float*)d_in[ii++];

  // ---- workspace arena (floats) ----
  float* wsf = (float*)d_ws;
  size_t off = 0;
  auto F = [&](size_t nf) { float* p = wsf + off; off += nf; return p; };
  float* pe400 = F(128 * 400);
  float* pe50  = F(128 * 50);
  float* Cx = F(8 * 128 * 400);
  float* Qx = F(8 * 128 * 50);
  float* Ce = F(8 * 128 * 400);
  float* Qe = F(8 * 128 * 50);
  float* M0 = F(409600); float* M1 = F(409600); float* M2 = F(409600); float* M3 = F(409600);
  float* e0 = F(409600); float* e1 = F(409600);
  float* tA = F(409600); float* tB = F(409600); float* tLn = F(409600); float* tDw = F(409600);
  float* tQ = F(409600); float* tK = F(409600); float* tV = F(409600); float* tH = F(409600);
  float* Ebuf    = F(8 * 400 * 128);
  float* rowterm = F(3200);
  float* S1 = F(8 * 400 * 50);
  float* S2 = F(8 * 400 * 50);
  float* At = F(409600);
  float* Bt = F(409600);
  float* X4   = F(8 * 512 * 400);
  float* X4dw = F(8 * 512 * 400);
  float* uni = F(13107200);   // union region: embed temps / attention S / cq T
  float* big1 = uni;                   // (8,64,L,16) char gather
  float* big0 = uni + 3276800;         // (8,64,L,16) dw2 out
  float* big2 = uni + 6553600;         // (8,128,L,16) pointwise out
  float* Sbuf = uni;                   // (64,L,L) attention scores
  float* Tbuf = uni;                   // (8,400,400) cq T
  float* embin  = X4;                  // (8,428,L) reuse (embedding phase only)
  float* embtmp = X4dw;

  // ---- positional encodings ----
  pe_k<<<(51200 + 255) / 256, 256, 0, st>>>(pe400, 400, 51200);
  pe_k<<<(6400 + 255) / 256, 256, 0, st>>>(pe50, 50, 6400);

  // ---- embedding ----
  auto embed = [&](float* dst, const int* wid, const int* cid, int L) {
    const int n1 = 8 * 64 * L * 16;
    gather_char_k<<<(n1 + 255) / 256, 256, 0, st>>>(cid, char_emb, big1, L, n1);
    dwconv2d_k<<<(n1 + 255) / 256, 256, 0, st>>>(big1, c2_dw, c2_dwb, big0, L, n1);
    { GemmArgs g = GA(c2_pw, 64, big0, L * 16, big2, 128, L * 16, 64);
      g.xStride = (long)64 * L * 16; g.oStr1 = (long)128 * L * 16; g.bias = c2_pwb;
      gemm(st, g, 8); }
    const int n2 = 8 * 128 * L;
    relumax_k<<<(n2 + 255) / 256, 256, 0, st>>>(big2, embin, L, n2);
    const int n3 = 8 * 300 * L;
    gather_word_k<<<(n3 + 255) / 256, 256, 0, st>>>(wid, word_emb, embin, L, n3);
    const int n4 = 8 * 428 * L;
    dwconv1d_k<<<(n4 + 255) / 256, 256, 0, st>>>(embin, c1_dw, c1_dwb, embtmp, 428, L, 5, n4);
    { GemmArgs g = GA(c1_pw, 428, embtmp, L, dst, 128, L, 428);
      g.xStride = (long)428 * L; g.oStr1 = (long)128 * L; g.bias = c1_pwb; gemm(st, g, 8); }
    for (int i = 0; i < 2; ++i) {
      { GemmArgs g = GA(gate_w + i * 16384, 128, dst, L, tLn, 128, L, 128);
        g.xStride = (long)128 * L; g.oStr1 = (long)128 * L; g.bias = gate_b + i * 128;
        g.act = 2; gemm(st, g, 8); }
      { GemmArgs g = GA(lin_w + i * 16384, 128, dst, L, tDw, 128, L, 128);
        g.xStride = (long)128 * L; g.oStr1 = (long)128 * L; g.bias = lin_b + i * 128;
        g.act = 1; gemm(st, g, 8); }
      highway_k<<<(n2 + 255) / 256, 256, 0, st>>>(dst, tLn, tDw, n2);
    }
  };
  embed(Cx, Cwid, Ccid, 400);
  embed(Qx, Qwid, Qcid, 50);

  // ---- context / question encoders ----
  enc_block(st, Ce, Cx, 400, cEnc, pe400, tA, tB, tLn, tDw, tQ, tK, tV, tH, Sbuf);
  enc_block(st, Qe, Qx, 50,  qEnc, pe50,  tA, tB, tLn, tDw, tQ, tK, tV, tH, Sbuf);

  // ---- context-query attention ----
  {
    const int nE = 8 * 400 * 128;
    cq_E_k<<<(nE + 255) / 256, 256, 0, st>>>(cqW, Ce, Ebuf, nE);
    cq_rowterm_k<<<3200, 128, 0, st>>>(cqW, Ce, rowterm);
    { GemmArgs g = GA(Ebuf, 128, Qe, 50, S1, 400, 50, 128);
      g.aStride = (long)400 * 128; g.xStride = (long)128 * 50; g.oStr1 = (long)400 * 50;
      g.rowBias = rowterm; gemm(st, g, 8); }
    copy_k<<<(160000 + 255) / 256, 256, 0, st>>>(S1, S2, 160000);
    rowsoftmax_k<<<dim3(400, 8), 256, 0, st>>>(S1, 400, 50);   // softmax over q
    colsoftmax_k<<<dim3(50, 8), 256, 0, st>>>(S2, 400, 50);    // softmax over c
    // A^T[d,c] = sum_q Qe[d,q] S1[c,q]
    { GemmArgs g = GA(Qe, 50, S1, 50, At, 128, 400, 50);
      g.aStride = (long)128 * 50; g.transX = 1; g.xStride = (long)400 * 50;
      g.oStr1 = (long)128 * 400; gemm(st, g, 8); }
    // T[c,j] = sum_q S1[c,q] S2[j,q]
    { GemmArgs g = GA(S1, 50, S2, 50, Tbuf, 400, 400, 50);
      g.aStride = (long)400 * 50; g.transX = 1; g.xStride = (long)400 * 50;
      g.oStr1 = (long)400 * 400; gemm(st, g, 8); }
    // Bm^T[d,c] = sum_j Ce[d,j] T[c,j]
    { GemmArgs g = GA(Ce, 400, Tbuf, 400, Bt, 128, 400, 400);
      g.aStride = (long)128 * 400; g.transX = 1; g.xStride = (long)400 * 400;
      g.oStr1 = (long)128 * 400; gemm(st, g, 8); }
    const int n4 = 8 * 512 * 400;
    cq_concat_k<<<(n4 + 255) / 256, 256, 0, st>>>(Ce, At, Bt, X4, n4);
    dwconv1d_k<<<(n4 + 255) / 256, 256, 0, st>>>(X4, rs_dw, nullptr, X4dw, 512, 400, 5, n4);
    { GemmArgs g = GA(rs_pw, 512, X4dw, 400, M0, 128, 400, 512);
      g.xStride = (long)512 * 400; g.oStr1 = (long)128 * 400; gemm(st, g, 8); }
  }

  // ---- model encoder stacks: M1, M2, M3 ----
  auto model_enc = [&](float* out, const float* in) {
    const float* s = in;
    for (int it = 0; it < 7; ++it) {
      float* d = (it == 6) ? out : ((it & 1) ? e1 : e0);
      enc_block(st, d, s, 400, mEnc, pe400, tA, tB, tLn, tDw, tQ, tK, tV, tH, Sbuf);
      s = d;
    }
  };
  model_enc(M1, M0);
  model_enc(M2, M1);
  model_enc(M3, M2);

  // ---- pointer heads ----
  float* out = (float*)d_out;
  const int np = 8 * 400;
  ptr_score_k<<<(np + 255) / 256, 256, 0, st>>>(ptr_W1, M1, M2, out, 400, np);
  ptr_score_k<<<(np + 255) / 256, 256, 0, st>>>(ptr_W2, M1, M3, out + 3200, 400, np);
  rowsoftmax_k<<<dim3(1, 8), 256, 0, st>>>(out, 1, 400);
  rowsoftmax_k<<<dim3(1, 8), 256, 0, st>>>(out + 3200, 1, 400);
}